// NODE_8083128451139
// MI455X (gfx1250) — compile-verified
//
#include <hip/hip_runtime.h>
#include <hip/hip_bf16.h>
#include <stdint.h>

// ---------------------------------------------------------------------------
// Problem constants (B=65536, D=1024, T=64, DEPTH=1, O=2)
// ---------------------------------------------------------------------------
#define BATCH  65536
#define D_DIM  1024
#define T_DIM  64
#define KSTEPS (D_DIM / 4)     // 256 k-steps of V_WMMA_F32_16X16X4_F32

typedef float v2f __attribute__((ext_vector_type(2)));
typedef float v8f __attribute__((ext_vector_type(8)));

// ---------------------------------------------------------------------------
// Prep kernel: pack feature_selectors [T=64][D=1024] (row-major over d) into
// the exact per-lane layout of the WMMA 4x16 f32 B fragment:
//   fragment fidx = s*4 + j   (s = k-step, j = 16-tree N tile)
//   lane L (hi = L>>4, ln = L&15):
//     VGPR0 = B[k0+2*hi][16j+ln] = fs[16j+ln][4s+2*hi]
//     VGPR1 = B[k0+2*hi+1][..]  = fs[16j+ln][4s+2*hi+1]
// Stored as one v2f per (fragment, lane): 256 fragments/K * 4 tiles * 32 lanes
// * 8B = 256 KiB in d_ws.  Hot-loop B read becomes one coalesced b64 per lane.
// ---------------------------------------------------------------------------
__global__ void pack_b_kernel(const float* __restrict__ fs,
                              float* __restrict__ pack) {
  int tid  = blockIdx.x * blockDim.x + threadIdx.x;   // 0..32767
  int s    = tid >> 7;          // k-step
  int rem  = tid & 127;
  int j    = rem >> 5;          // N tile (16 trees)
  int lane = rem & 31;
  int hi   = lane >> 4;
  int ln   = lane & 15;
  int t  = 16 * j + ln;         // tree index
  int d0 = 4 * s + 2 * hi;      // feature index
  v2f v;
  v.x = fs[t * D_DIM + d0];
  v.y = fs[t * D_DIM + d0 + 1];
  ((v2f*)pack)[tid] = v;
}

// ---------------------------------------------------------------------------
// Main kernel: 256 blocks x 256 threads (8 waves). Each wave owns 32 rows of x
// (2 M-tiles) x all 64 trees (4 N-tiles) => 8 f32 16x16 accumulators.
// Packed B (256 KiB) is async-staged into LDS once per block, then re-read
// with conflict-free ds_load_b64.  x is streamed once with nontemporal loads.
// Epilogue fuses sigmoid + leaf-weight mixing + tree reduction.
// ---------------------------------------------------------------------------
__global__ void __launch_bounds__(256, 1)
forest_kernel(const float* __restrict__ x,
              const float* __restrict__ thresholds,    // [T]
              const float* __restrict__ leaf_weights,  // [T][2][2]
              const float* __restrict__ packB,         // 256 KiB fragments
              float* __restrict__ out) {               // [B][2]
  extern __shared__ float ldsb[];                      // 262144 bytes dynamic

  const int lane = threadIdx.x & 31;
  const int wave = threadIdx.x >> 5;
  const int hi   = lane >> 4;
  const int ln   = lane & 15;

  // ---- Stage packed B into LDS via CDNA5 async global->LDS DMA ----------
  {
    unsigned lds_base = (unsigned)(uintptr_t)ldsb;     // flat LDS addr[31:0]
    for (int i = threadIdx.x; i < 16384; i += 256) {   // 16384 * 16B = 256 KiB
      unsigned lds_addr = lds_base + i * 16;
      unsigned goff     = i * 16;                      // GVS: saddr + vgpr off
      asm volatile("global_load_async_to_lds_b128 %0, %1, %2"
                   :
                   : "v"(lds_addr), "v"(goff), "s"(packB)
                   : "memory");
    }
    asm volatile("s_wait_asynccnt 0" ::: "memory");
  }
  __syncthreads();

  // ---- Main GEMM loop -----------------------------------------------------
  const int tile_row = (blockIdx.x * 8 + wave) * 32;   // 32 rows per wave
  // A fragment source: lane ln streams row tile_row+ln (tile 0) and +16
  // (tile 1); the 16x4 f32 A layout wants K pair {2*hi, 2*hi+1} per lane.
  const float* xr0 = x + (size_t)(tile_row + ln) * D_DIM + 2 * hi;
  const float* xr1 = xr0 + 16 * D_DIM;

  v8f acc[2][4] = {};

#pragma unroll 4
  for (int s = 0; s < KSTEPS; ++s) {
    v2f a0 = __builtin_nontemporal_load((const v2f*)(xr0 + 4 * s));
    v2f a1 = __builtin_nontemporal_load((const v2f*)(xr1 + 4 * s));
    const v2f* bp = ((const v2f*)ldsb) + (s * 4) * 32 + lane;
#pragma unroll
    for (int j = 0; j < 4; ++j) {
      v2f b = bp[j * 32];                              // ds_load_b64, no bank conflicts
      acc[0][j] = __builtin_amdgcn_wmma_f32_16x16x4_f32(
          false, a0, false, b, (short)0, acc[0][j], false, false);
      acc[1][j] = __builtin_amdgcn_wmma_f32_16x16x4_f32(
          false, a1, false, b, (short)0, acc[1][j], false, false);
    }
  }

  // ---- Fused epilogue: sigmoid + leaf mixing + tree reduction -------------
  // C/D layout: VGPR r, lanes ln (half hi): element = logits[row=tile_row +
  // 16*mt + r + 8*hi][tree = 16*j + ln].
  float  thr_l[4];
  float4 w_l[4];
#pragma unroll
  for (int j = 0; j < 4; ++j) {
    int t    = 16 * j + ln;
    thr_l[j] = thresholds[t];
    w_l[j]   = ((const float4*)leaf_weights)[t];  // {W[t][0][0],W[t][0][1],W[t][1][0],W[t][1][1]}
  }

#pragma unroll
  for (int mt = 0; mt < 2; ++mt) {
#pragma unroll
    for (int r = 0; r < 8; ++r) {
      float s0 = 0.0f, s1 = 0.0f;
#pragma unroll
      for (int j = 0; j < 4; ++j) {
        float z = acc[mt][j][r] - thr_l[j];
        // fast sigmoid: v_exp_f32 + v_rcp_f32 (no IEEE divide chain)
        float p = __builtin_amdgcn_rcpf(1.0f + __expf(-z));
        // p*W0 + (1-p)*W1 = p*(W0-W1) + W1
        s0 = fmaf(p, w_l[j].x - w_l[j].z, s0 + w_l[j].z);
        s1 = fmaf(p, w_l[j].y - w_l[j].w, s1 + w_l[j].w);
      }
      // reduce the 16 tree-partials held across each 16-lane half
      for (int m = 8; m >= 1; m >>= 1) {
        s0 += __shfl_xor(s0, m, 16);
        s1 += __shfl_xor(s1, m, 16);
      }
      if (ln == 0) {
        int row = tile_row + 16 * mt + r + 8 * hi;
        ((float2*)out)[row] = make_float2(s0, s1);
      }
    }
  }
}

// ---------------------------------------------------------------------------
// Launch
// ---------------------------------------------------------------------------
extern "C" void kernel_launch(void* const* d_in, const int* in_sizes, int n_in,
                              void* d_out, int out_size, void* d_ws, size_t ws_size,
                              hipStream_t stream) {
  const float* x   = (const float*)d_in[0];   // [65536,1024]
  const float* fs  = (const float*)d_in[1];   // [64,1,1024]
  const float* thr = (const float*)d_in[2];   // [64,1]
  const float* lw  = (const float*)d_in[3];   // [64,2,2]
  float*       out = (float*)d_out;           // [65536,2]
  float*       pk  = (float*)d_ws;            // 256 KiB packed B fragments

  (void)in_sizes; (void)n_in; (void)out_size; (void)ws_size;

  hipFuncSetAttribute((const void*)forest_kernel,
                      hipFuncAttributeMaxDynamicSharedMemorySize, 262144);

  pack_b_kernel<<<128, 256, 0, stream>>>(fs, pk);
  forest_kernel<<<256, 256, 262144, stream>>>(x, thr, lw, pk, out);
}